// YOLOv1Loss_23854248362731
// MI455X (gfx1250) — compile-verified
//
#include <hip/hip_runtime.h>
#include <hip/hip_bf16.h>
#include <math.h>

#define PRED_STRIDE 11   // B*5+C = 2*5+1
#define TGT_STRIDE  6    // 4+1+1
#define N_CELLS     (32768*7*7)   // 1,605,632
#define TPB         256
#define TILE        256
#define NTILES      (N_CELLS/TILE) // 6272, exact
#define NBLOCKS     1024
#define EPSF        1e-6f

typedef __attribute__((ext_vector_type(2))) float v2f;
typedef __attribute__((ext_vector_type(8))) float v8f;

// ---------- CDNA5 async global->LDS with portable fallback ----------
#if defined(__has_builtin)
#if __has_builtin(__builtin_amdgcn_global_load_async_to_lds_b32)
#define HAVE_ASYNC_LDS 1
#endif
#endif

__device__ __forceinline__ void async_copy_f32(const float* g, float* l) {
#if defined(HAVE_ASYNC_LDS)
  __builtin_amdgcn_global_load_async_to_lds_b32(
      (__attribute__((address_space(1))) int*)(g),
      (__attribute__((address_space(3))) int*)(l), 0, 0);
#else
  *l = *g;  // sync fallback (regular load + ds_store)
#endif
}

__device__ __forceinline__ void wait_async_prev17() {
#if defined(HAVE_ASYNC_LDS)
#if __has_builtin(__builtin_amdgcn_s_wait_asynccnt)
  __builtin_amdgcn_s_wait_asynccnt(17);
#else
  asm volatile("s_wait_asynccnt 17" ::: "memory");
#endif
#endif
}

__device__ __forceinline__ void wait_async_all() {
#if defined(HAVE_ASYNC_LDS)
#if __has_builtin(__builtin_amdgcn_s_wait_asynccnt)
  __builtin_amdgcn_s_wait_asynccnt(0);
#else
  asm volatile("s_wait_asynccnt 0" ::: "memory");
#endif
#endif
}

// ---------- 32-lane f32 sum via V_WMMA_F32_16X16X4_F32 ----------
// A (16x4 f32, documented layout): VGPR0 = {lanes0-15:K0, lanes16-31:K2},
// VGPR1 = K1/K3. Put x in slot0, 0 in slot1 -> row m = x[lane m] + x[lane m+16].
// B = all ones (any layout, all slots 1.0) -> D[m][n] = rowsum[m].
// D layout: VGPR v: lanes0-15 -> (M=v, N=lane); lanes16-31 -> (M=v+8, N=lane-16).
// Per-lane sum of the 8 D VGPRs = sum of 8 rows of its column half; adding the
// lane+16 value (other 8 rows) gives the full 32-lane sum (valid in lanes 0-15).
__device__ __forceinline__ float wave_sum_wmma(float x) {
  v2f a; a[0] = x;    a[1] = 0.0f;
  v2f b; b[0] = 1.0f; b[1] = 1.0f;
  v8f c = {};
  v8f d = __builtin_amdgcn_wmma_f32_16x16x4_f32(
      /*neg_a=*/false, a, /*neg_b=*/false, b,
      /*c_mod=*/(short)0, c, /*reuse_a=*/false, /*reuse_b=*/false);
  float s = d[0] + d[1] + d[2] + d[3] + d[4] + d[5] + d[6] + d[7];
  s += __shfl_down(s, 16, 32);
  return s;  // total in lane 0
}

// ---------- main streaming kernel: per-block partial sums of 4 components ----
__global__ __launch_bounds__(TPB) void yolo_main(
    const float* __restrict__ pred, const float* __restrict__ tgt,
    float* __restrict__ ws) {
  __shared__ float sp[2][TILE * PRED_STRIDE];  // 2 x 11264 B
  __shared__ float st[2][TILE * TGT_STRIDE];   // 2 x  6144 B
  __shared__ float spart[(TPB / 32) * 4];

  const int t = threadIdx.x;
  float a_coord = 0.0f, a_cobj = 0.0f, a_cnoobj = 0.0f, a_cls = 0.0f;

  auto issue = [&](int buf, int tl) {
    const float* gp = pred + (size_t)tl * (TILE * PRED_STRIDE);
    const float* gq = tgt  + (size_t)tl * (TILE * TGT_STRIDE);
#pragma unroll
    for (int k = 0; k < PRED_STRIDE; ++k)
      async_copy_f32(gp + k * TILE + t, &sp[buf][k * TILE + t]);
#pragma unroll
    for (int k = 0; k < TGT_STRIDE; ++k)
      async_copy_f32(gq + k * TILE + t, &st[buf][k * TILE + t]);
  };

  int tile = blockIdx.x;
  int buf = 0;
  if (tile < NTILES) issue(0, tile);

  while (tile < NTILES) {
    const int nxt = tile + NBLOCKS;
    if (nxt < NTILES) { issue(buf ^ 1, nxt); wait_async_prev17(); }
    else              { wait_async_all(); }
    __syncthreads();

    const float* p = &sp[buf][t * PRED_STRIDE];
    const float* q = &st[buf][t * TGT_STRIDE];

    const float tx = q[0], ty = q[1], tw = q[2], th = q[3];
    const float tcf = q[4], tcl = q[5];
    const float obj   = (tcf == 1.0f) ? 1.0f : 0.0f;
    const float noobj = (tcf == 0.0f) ? 1.0f : 0.0f;
    const float t1x = tx - 0.5f * tw, t1y = ty - 0.5f * th;
    const float t2x = tx + 0.5f * tw, t2y = ty + 0.5f * th;
    const float tarea = tw * th;

    float sx[2], sy[2], sc[2], aw[2], ah[2], iou[2];
#pragma unroll
    for (int b = 0; b < 2; ++b) {
      const float px = p[b * 5 + 0], py = p[b * 5 + 1];
      aw[b] = fabsf(p[b * 5 + 2]);
      ah[b] = fabsf(p[b * 5 + 3]);
      sx[b] = 1.0f / (1.0f + expf(-px));
      sy[b] = 1.0f / (1.0f + expf(-py));
      sc[b] = 1.0f / (1.0f + expf(-p[b * 5 + 4]));
      const float p1x = sx[b] - 0.5f * aw[b], p1y = sy[b] - 0.5f * ah[b];
      const float p2x = sx[b] + 0.5f * aw[b], p2y = sy[b] + 0.5f * ah[b];
      const float lox = fmaxf(p1x, t1x), loy = fmaxf(p1y, t1y);
      const float hix = fminf(p2x, t2x), hiy = fminf(p2y, t2y);
      const float inter = fmaxf(hix - lox, 0.0f) * fmaxf(hiy - loy, 0.0f);
      const float parea = aw[b] * ah[b];  // |p2-p1| == (aw,ah)
      iou[b] = inter / (parea + tarea - inter + EPSF);
    }
    const int best = (iou[1] > iou[0]) ? 1 : 0;  // argmax, first wins ties

    const float dx = sx[best] - tx, dy = sy[best] - ty;
    const float sw = sqrtf(aw[best] + EPSF) - sqrtf(tw + EPSF);
    const float sh = sqrtf(ah[best] + EPSF) - sqrtf(th + EPSF);
    a_coord += 5.0f * obj * (dx * dx + dy * dy + sw * sw + sh * sh);

    const float dc = sc[best] - tcf;
    a_cobj += obj * dc * dc;

    const float d0 = sc[0] - tcf, d1 = sc[1] - tcf;
    a_cnoobj += 0.5f * noobj * (d0 * d0 + d1 * d1);

    const float pc = p[10];
    const float bce = fmaxf(pc, 0.0f) - pc * tcl + log1pf(expf(-fabsf(pc)));
    a_cls += obj * bce;

    __syncthreads();  // all waves done reading buf before it is refilled
    buf ^= 1;
    tile = nxt;
  }

  // wave32 reduction via WMMA adder tree, then 8-wave combine
  const float r0 = wave_sum_wmma(a_coord);
  const float r1 = wave_sum_wmma(a_cobj);
  const float r2 = wave_sum_wmma(a_cnoobj);
  const float r3 = wave_sum_wmma(a_cls);
  const int wave = t >> 5, lane = t & 31;
  if (lane == 0) {
    spart[wave * 4 + 0] = r0; spart[wave * 4 + 1] = r1;
    spart[wave * 4 + 2] = r2; spart[wave * 4 + 3] = r3;
  }
  __syncthreads();
  if (t == 0) {
    float c0 = 0, c1 = 0, c2 = 0, c3 = 0;
#pragma unroll
    for (int w = 0; w < TPB / 32; ++w) {
      c0 += spart[w * 4 + 0]; c1 += spart[w * 4 + 1];
      c2 += spart[w * 4 + 2]; c3 += spart[w * 4 + 3];
    }
    float* o = ws + (size_t)blockIdx.x * 4;
    o[0] = c0; o[1] = c1; o[2] = c2; o[3] = c3;
  }
}

// ---------- deterministic final reduction of 1024 block partials ----------
__global__ __launch_bounds__(256) void yolo_final(
    const float* __restrict__ ws, float* __restrict__ out) {
  __shared__ float spart[(256 / 32) * 4];
  const int t = threadIdx.x;
  float c0 = 0, c1 = 0, c2 = 0, c3 = 0;
  for (int s = t; s < NBLOCKS; s += 256) {
    const float* w = ws + (size_t)s * 4;
    c0 += w[0]; c1 += w[1]; c2 += w[2]; c3 += w[3];
  }
  c0 = wave_sum_wmma(c0);
  c1 = wave_sum_wmma(c1);
  c2 = wave_sum_wmma(c2);
  c3 = wave_sum_wmma(c3);
  const int wave = t >> 5, lane = t & 31;
  if (lane == 0) {
    spart[wave * 4 + 0] = c0; spart[wave * 4 + 1] = c1;
    spart[wave * 4 + 2] = c2; spart[wave * 4 + 3] = c3;
  }
  __syncthreads();
  if (t == 0) {
    float coord = 0, cobj = 0, cnoobj = 0, cls = 0;
#pragma unroll
    for (int w = 0; w < 8; ++w) {
      coord  += spart[w * 4 + 0]; cobj += spart[w * 4 + 1];
      cnoobj += spart[w * 4 + 2]; cls  += spart[w * 4 + 3];
    }
    const float inv = 1.0f / 32768.0f;
    out[0] = (coord + cobj + cnoobj + cls) * inv;
    out[1] = coord * inv;
    out[2] = cobj * inv;
    out[3] = cnoobj * inv;
    out[4] = cls * inv;
  }
}

extern "C" void kernel_launch(void* const* d_in, const int* in_sizes, int n_in,
                              void* d_out, int out_size, void* d_ws, size_t ws_size,
                              hipStream_t stream) {
  const float* pred = (const float*)d_in[0];  // predictions (BATCH,7,7,11)
  const float* tgt  = (const float*)d_in[1];  // targets     (BATCH,7,7,6)
  float* out = (float*)d_out;                 // 5 scalars
  float* ws  = (float*)d_ws;                  // NBLOCKS*4 floats of partials

  yolo_main<<<NBLOCKS, TPB, 0, stream>>>(pred, tgt, ws);
  yolo_final<<<1, 256, 0, stream>>>(ws, out);
}